// GCNEncoder_24318104830702
// MI455X (gfx1250) — compile-verified
//
#include <hip/hip_runtime.h>
#include <hip/hip_bf16.h>

typedef float v2f __attribute__((ext_vector_type(2)));
typedef float v4f __attribute__((ext_vector_type(4)));
typedef float v8f __attribute__((ext_vector_type(8)));

// ---------------------------------------------------------------------------
// Guaranteed-hardware non-returning f32 atomic add, device scope.
// VGLOBAL GLOBAL_ATOMIC_ADD_F32 (opcode 86): resolves in L2 atomic units.
// ---------------------------------------------------------------------------
__device__ __forceinline__ void atomic_add_f32_dev(float* p, float v) {
    asm volatile("global_atomic_add_f32 %0, %1, off scope:SCOPE_DEV"
                 :: "v"(p), "v"(v)
                 : "memory");
}

// ---------------------------------------------------------------------------
// Degree / dinv kernels
// ---------------------------------------------------------------------------
__global__ __launch_bounds__(256) void init_deg_kernel(float* __restrict__ deg, int n) {
    int i = blockIdx.x * blockDim.x + threadIdx.x;
    if (i < n) deg[i] = 1.0f;  // self-loop contribution
}

__global__ __launch_bounds__(256) void deg_scatter_kernel(const long long* __restrict__ dst,
                                                          float* __restrict__ deg, int nE) {
    int e = blockIdx.x * blockDim.x + threadIdx.x;
    if (e < nE) {
        int d = (int)dst[e];
        atomic_add_f32_dev(&deg[d], 1.0f);
    }
}

__global__ __launch_bounds__(256) void deg_to_dinv_kernel(float* __restrict__ deg, int n) {
    int i = blockIdx.x * blockDim.x + threadIdx.x;
    if (i < n) deg[i] = rsqrtf(deg[i]);
}

// ---------------------------------------------------------------------------
// Transpose W[K,N] -> Wt[N,K] so GEMM B fragments become contiguous float2.
// W is at most 128x128 = 64KB -> trivial, stays L2/L0 hot.
// ---------------------------------------------------------------------------
__global__ __launch_bounds__(256) void transpose_kernel(const float* __restrict__ W,
                                                        float* __restrict__ Wt,
                                                        int K, int N) {
    int i = blockIdx.x * blockDim.x + threadIdx.x;
    if (i < K * N) {
        int k = i / N, n = i % N;
        Wt[(size_t)n * K + k] = W[i];
    }
}

// ---------------------------------------------------------------------------
// f32 WMMA GEMM:  C[M,N] = A[M,K] @ W[K,N], W given transposed as Wt[N,K].
// Row-major, M%16==0, K%4==0, N == NTILES*16.
// One wave computes one 16-row strip across all N tiles using
// V_WMMA_F32_16X16X4_F32 (full fp32 precision, matches reference).
// A frag: lanes 0-15 rows 0-15 K={k,k+1}; lanes 16-31 rows 0-15 K={k+2,k+3}.
// B frag: same half-wave K split, col = lane%16  -> one b64 load from Wt.
// ---------------------------------------------------------------------------
template <int NTILES>
__global__ __launch_bounds__(256) void gemm_wmma_f32(const float* __restrict__ A,
                                                     const float* __restrict__ Wt,
                                                     float* __restrict__ C,
                                                     int Mtiles, int K) {
    const int N = NTILES * 16;
    int tile = blockIdx.x * (blockDim.x >> 5) + (threadIdx.x >> 5);
    if (tile >= Mtiles) return;             // whole-wave exit: EXEC stays all-1s for WMMA

    const int lane = threadIdx.x & 31;
    const int half = lane >> 4;             // 0: K pair {0,1}; 1: K pair {2,3}
    const int l    = lane & 15;
    const int m0   = tile * 16;

    v8f acc[NTILES] = {};

    for (int k = 0; k < K; k += 4) {
        v2f a = *(const v2f*)(A + (size_t)(m0 + l) * K + k + 2 * half);
#pragma unroll
        for (int t = 0; t < NTILES; ++t) {
            v2f b = *(const v2f*)(Wt + (size_t)(t * 16 + l) * K + k + 2 * half);
            acc[t] = __builtin_amdgcn_wmma_f32_16x16x4_f32(
                /*neg_a=*/false, a, /*neg_b=*/false, b,
                /*c_mod=*/(short)0, acc[t], /*reuse_a=*/false, /*reuse_b=*/false);
        }
    }

    // C/D layout: VGPR v, lanes 0-15 -> row m0+v; lanes 16-31 -> row m0+8+v
#pragma unroll
    for (int t = 0; t < NTILES; ++t) {
#pragma unroll
        for (int v = 0; v < 8; ++v) {
            C[(size_t)(m0 + v + 8 * half) * N + t * 16 + l] = acc[t][v];
        }
    }
}

// ---------------------------------------------------------------------------
// agg init: agg[i, c] = h[i, c] * dinv[i]^2 + bias[c]   (self-loop + bias)
// ---------------------------------------------------------------------------
template <int CH>
__global__ __launch_bounds__(256) void init_agg_kernel(const float* __restrict__ h,
                                                       const float* __restrict__ dinv,
                                                       const float* __restrict__ bias,
                                                       float* __restrict__ agg, int nNodes) {
    long long i = (long long)blockIdx.x * blockDim.x + threadIdx.x;
    long long total = (long long)nNodes * CH;
    if (i < total) {
        int node = (int)(i / CH);
        int ch   = (int)(i % CH);
        float di = dinv[node];
        agg[i] = h[i] * di * di + bias[ch];
    }
}

// ---------------------------------------------------------------------------
// Edge scatter: one wave per edge. agg[dst] += h[src] * dinv[src]*dinv[dst].
// Lane carries CH/32 contiguous channels; hardware f32 atomics (L2-resident).
// ---------------------------------------------------------------------------
template <int CH>
__global__ __launch_bounds__(256) void edge_scatter_kernel(const float* __restrict__ h,
                                                           const float* __restrict__ dinv,
                                                           const long long* __restrict__ src,
                                                           const long long* __restrict__ dst,
                                                           float* __restrict__ agg, int nE) {
    int e = blockIdx.x * (blockDim.x >> 5) + (threadIdx.x >> 5);
    if (e >= nE) return;
    const int lane = threadIdx.x & 31;
    const int s = (int)src[e];
    const int d = (int)dst[e];
    const float c = dinv[s] * dinv[d];

    constexpr int V = CH / 32;  // 4 (128ch) or 2 (64ch) floats per lane
    const float* hp = h   + (size_t)s * CH + lane * V;
    float*       op = agg + (size_t)d * CH + lane * V;

    if constexpr (V == 4) {
        v4f hv = *(const v4f*)hp;
#pragma unroll
        for (int i = 0; i < 4; ++i)
            atomic_add_f32_dev(op + i, hv[i] * c);
    } else {
        v2f hv = *(const v2f*)hp;
#pragma unroll
        for (int i = 0; i < V; ++i)
            atomic_add_f32_dev(op + i, hv[i] * c);
    }
}

// ---------------------------------------------------------------------------
// In-place ReLU
// ---------------------------------------------------------------------------
__global__ __launch_bounds__(256) void relu_kernel(float* __restrict__ x, long long n) {
    long long i = (long long)blockIdx.x * blockDim.x + threadIdx.x;
    if (i < n) x[i] = fmaxf(x[i], 0.0f);
}

// ---------------------------------------------------------------------------
// Launch
// ---------------------------------------------------------------------------
extern "C" void kernel_launch(void* const* d_in, const int* in_sizes, int n_in,
                              void* d_out, int out_size, void* d_ws, size_t ws_size,
                              hipStream_t stream) {
    constexpr int IN_CH = 128, HID_CH = 128, OUT_CH = 64;

    const float*     x   = (const float*)d_in[0];
    const long long* ei  = (const long long*)d_in[1];   // int64 edge_index [2, E]
    const float*     W1  = (const float*)d_in[2];
    const float*     b1  = (const float*)d_in[3];
    const float*     W2  = (const float*)d_in[4];
    const float*     b2  = (const float*)d_in[5];
    float*           out = (float*)d_out;

    const int nNodes = in_sizes[0] / IN_CH;   // 100000
    const int nE     = in_sizes[1] / 2;       // 1600000
    const long long* src = ei;                // edge_index[0]
    const long long* dst = ei + nE;           // edge_index[1]

    // Workspace (floats): dinv | Wt (128*128) | h (nNodes*128, reused for h2) | agg1
    float* ws    = (float*)d_ws;
    float* dinv  = ws;
    float* Wt    = ws + (((size_t)nNodes + 63) / 64) * 64;
    float* h     = Wt + (size_t)IN_CH * HID_CH;
    float* agg1  = h + (size_t)nNodes * HID_CH;

    const int TPB = 256;
    const int wavesPerBlock = TPB / 32;

    // --- degrees (shared by both layers) ---
    init_deg_kernel<<<(nNodes + TPB - 1) / TPB, TPB, 0, stream>>>(dinv, nNodes);
    deg_scatter_kernel<<<(nE + TPB - 1) / TPB, TPB, 0, stream>>>(dst, dinv, nE);
    deg_to_dinv_kernel<<<(nNodes + TPB - 1) / TPB, TPB, 0, stream>>>(dinv, nNodes);

    const int Mtiles     = nNodes / 16;                              // 6250
    const int gemmBlocks = (Mtiles + wavesPerBlock - 1) / wavesPerBlock;

    // --- layer 1: h = x @ W1 ; agg1 = self + bias ; scatter ; relu ---
    transpose_kernel<<<(IN_CH * HID_CH + TPB - 1) / TPB, TPB, 0, stream>>>(W1, Wt, IN_CH, HID_CH);
    gemm_wmma_f32<HID_CH / 16><<<gemmBlocks, TPB, 0, stream>>>(x, Wt, h, Mtiles, IN_CH);

    long long n1 = (long long)nNodes * HID_CH;
    init_agg_kernel<HID_CH><<<(int)((n1 + TPB - 1) / TPB), TPB, 0, stream>>>(h, dinv, b1, agg1, nNodes);
    edge_scatter_kernel<HID_CH><<<(nE + wavesPerBlock - 1) / wavesPerBlock, TPB, 0, stream>>>(
        h, dinv, src, dst, agg1, nE);
    relu_kernel<<<(int)((n1 + TPB - 1) / TPB), TPB, 0, stream>>>(agg1, n1);

    // --- layer 2: h2 = relu(agg1) @ W2 (reuse h buffer) ; out = self + bias ; scatter ---
    transpose_kernel<<<(HID_CH * OUT_CH + TPB - 1) / TPB, TPB, 0, stream>>>(W2, Wt, HID_CH, OUT_CH);
    gemm_wmma_f32<OUT_CH / 16><<<gemmBlocks, TPB, 0, stream>>>(agg1, Wt, h, Mtiles, HID_CH);

    long long n2 = (long long)nNodes * OUT_CH;
    init_agg_kernel<OUT_CH><<<(int)((n2 + TPB - 1) / TPB), TPB, 0, stream>>>(h, dinv, b2, out, nNodes);
    edge_scatter_kernel<OUT_CH><<<(nE + wavesPerBlock - 1) / wavesPerBlock, TPB, 0, stream>>>(
        h, dinv, src, dst, out, nE);
}